// CDFNet_71579924955646
// MI455X (gfx1250) — compile-verified
//
#include <hip/hip_runtime.h>
#include <hip/hip_bf16.h>

// ---------------- problem constants ----------------
#define DD    16
#define NN    512
#define MH    5
#define BATCH 1024

// ---------------- workspace layout (floats) ----------------
// softplus_b10 weights + tanh(a) precomputed, normalized mixture, and L[m,n]
static constexpr int O_WN0 = 0;                    // 16*512*5   = 40960
static constexpr int O_WN1 = 40960;                // 16*512*25  = 204800
static constexpr int O_WN2 = 245760;               // 204800
static constexpr int O_WN3 = 450560;               // 40960
static constexpr int O_TA0 = 491520;               // 40960
static constexpr int O_TA1 = 532480;               // 40960
static constexpr int O_TA2 = 573440;               // 40960
static constexpr int O_AM  = 614400;               // 512
static constexpr int O_L   = 614912;               // 1024*512 = 524288
// total = 1139200 floats = ~4.56 MB of d_ws

typedef float v2f __attribute__((ext_vector_type(2)));
typedef float v8f __attribute__((ext_vector_type(8)));

// ---------------- fast math helpers ----------------
__device__ __forceinline__ float fast_tanh(float x) {
#if __has_builtin(__builtin_amdgcn_tanhf)
  return __builtin_amdgcn_tanhf(x);   // v_tanh_f32 (gfx1250 trans op)
#else
  float xc = fminf(fmaxf(x, -15.0f), 15.0f);
  float e = __expf(2.0f * xc);
  return (e - 1.0f) / (e + 1.0f);
#endif
}

__device__ __forceinline__ float fast_rcp(float x) {
#if __has_builtin(__builtin_amdgcn_rcpf)
  return __builtin_amdgcn_rcpf(x);    // single v_rcp_f32, no div-fixup chain
#else
  return 1.0f / x;
#endif
}

__device__ __forceinline__ float softplus_b10(float x) {
  float bx = 10.0f * x;
  float sp = (bx > 25.0f) ? bx : log1pf(__expf(bx));
  return 0.1f * sp;
}

// broadcast: every lane receives value held by lane (lane & 15)
__device__ __forceinline__ float perm_lo16(float v, int lane) {
  int idx = (lane & 15) << 2;
  return __int_as_float(__builtin_amdgcn_ds_bpermute(idx, __float_as_int(v)));
}

// ---------------- 5x5 (padded to 16x8x16) matmul via f32 WMMA ----------------
// Computes D[l, m] = sum_k wn[k, l] * x_k[m] + Cinit[l, m]
// x[k] holds x_k[m] in lane m (valid lanes 0..15). aw are per-lane A values:
//   aw0 = A[row, hi?2:0], aw1 = A[row, hi?3:1], aw2 = A[row, 4] (lo half only)
__device__ __forceinline__ v8f mm5(float aw0, float aw1, float aw2,
                                   const float x[MH], v8f cinit, int lane) {
  bool lo = lane < 16;
  float p2 = perm_lo16(x[2], lane);
  float p3 = perm_lo16(x[3], lane);
  v2f A;  A[0] = aw0;  A[1] = aw1;
  v2f B;  B[0] = lo ? x[0] : p2;  B[1] = lo ? x[1] : p3;
  v8f acc = __builtin_amdgcn_wmma_f32_16x16x4_f32(
      false, A, false, B, (short)0, cinit, false, false);
  v2f A2; A2[0] = aw2;           A2[1] = 0.0f;
  v2f B2; B2[0] = lo ? x[4] : 0.0f; B2[1] = 0.0f;
  acc = __builtin_amdgcn_wmma_f32_16x16x4_f32(
      false, A2, false, B2, (short)0, acc, false, false);
  return acc;
}

// One 5->5 layer: updates phi[] (activations) and pdv[] (derivative chain).
// wl: softplus'd weights (k*5+l), bl: bias, tal: tanh(a). All (d,n)-uniform ptrs.
__device__ __forceinline__ void layer55(const float* __restrict__ wl,
                                        const float* __restrict__ bl,
                                        const float* __restrict__ tal,
                                        float phi[MH], float pdv[MH], int lane) {
  int row = lane & 15;
  bool hi = lane >= 16;
  bool lo = !hi;
  // A = wn^T : A[l, k] = wn[k, l]
  float aw0 = (row < MH) ? wl[(hi ? 2 : 0) * MH + row] : 0.0f;
  float aw1 = (row < MH) ? wl[(hi ? 3 : 1) * MH + row] : 0.0f;
  float aw2 = (lo && row < MH) ? wl[4 * MH + row] : 0.0f;

  v8f cb, zc;
#pragma unroll
  for (int j = 0; j < 8; ++j) {
    cb[j] = (j < MH) ? (lo ? bl[j] : 0.0f) : 0.0f;
    zc[j] = 0.0f;
  }

  v8f z8  = mm5(aw0, aw1, aw2, phi, cb, lane);   // z^T   = wn^T phi^T + b
  v8f pd8 = mm5(aw0, aw1, aw2, pdv, zc, lane);   // (pd . wn)^T

#pragma unroll
  for (int l = 0; l < MH; ++l) {
    float z  = z8[l];
    float tz = fast_tanh(z);
    float ta = tal[l];
    float gate = 1.0f + ta * (1.0f - tz * tz);
    phi[l] = z + tz * ta;
    pdv[l] = pd8[l] * gate;
  }
}

// ---------------- kernels ----------------

// mode 0: softplus(beta=10); mode 1: tanh
__global__ __launch_bounds__(256) void xform_kernel(const float* __restrict__ src,
                                                    float* __restrict__ dst,
                                                    int count, int mode) {
  int i = blockIdx.x * blockDim.x + threadIdx.x;
  if (i >= count) return;
  float x = src[i];
  dst[i] = (mode == 0) ? softplus_b10(x) : fast_tanh(x);
}

// softplus(a_mix) normalized to sum 1 (one block of 512 threads)
__global__ __launch_bounds__(NN) void amix_kernel(const float* __restrict__ a_mix,
                                                  float* __restrict__ amn) {
  __shared__ float red[NN];
  int t = threadIdx.x;
  float x = a_mix[t];
  float sp = (x > 25.0f) ? x : log1pf(__expf(x));
  red[t] = sp;
  __syncthreads();
  for (int s = NN / 2; s > 0; s >>= 1) {
    if (t < s) red[t] += red[t + s];
    __syncthreads();
  }
  amn[t] = sp * fast_rcp(red[0]);
}

// Main MLP: one wave handles (component n, batch tile of 16), loops over d.
// Emits L[m, n] = sum_d log(phidot(m,d,n) + 1e-10)
__global__ __launch_bounds__(256) void mlp_kernel(const float* __restrict__ X,
                                                  const float* __restrict__ b0g,
                                                  const float* __restrict__ b1g,
                                                  const float* __restrict__ b2g,
                                                  const float* __restrict__ b3g,
                                                  const float* __restrict__ ws,
                                                  float* __restrict__ Lout) {
  int lane = threadIdx.x & 31;
  int wid  = blockIdx.x * (blockDim.x >> 5) + (threadIdx.x >> 5);
  int n  = wid >> 6;          // [0, 512)
  int bt = wid & 63;          // [0, 64) batch tiles of 16
  int mrow = (bt << 4) + (lane & 15);

  float Lsum = 0.0f;
#pragma unroll 1
  for (int d = 0; d < DD; ++d) {
    int dn = d * NN + n;
    const float* wn0 = ws + O_WN0 + dn * MH;
    const float* wn1 = ws + O_WN1 + dn * (MH * MH);
    const float* wn2 = ws + O_WN2 + dn * (MH * MH);
    const float* wn3 = ws + O_WN3 + dn * MH;
    const float* ta0 = ws + O_TA0 + dn * MH;
    const float* ta1 = ws + O_TA1 + dn * MH;
    const float* ta2 = ws + O_TA2 + dn * MH;
    const float* b0p = b0g + dn * MH;
    const float* b1p = b1g + dn * MH;
    const float* b2p = b2g + dn * MH;

    float x = X[mrow * DD + d];

    float phi[MH], pdv[MH];
    // layer 0: 1 -> 5 (scalar in K, pure VALU)
#pragma unroll
    for (int l = 0; l < MH; ++l) {
      float w  = wn0[l];
      float z  = fmaf(w, x, b0p[l]);
      float tz = fast_tanh(z);
      float ta = ta0[l];
      float gate = 1.0f + ta * (1.0f - tz * tz);
      phi[l] = z + tz * ta;
      pdv[l] = w * gate;         // phidots started at 1
    }
    // layers 1,2: 5 -> 5 via V_WMMA_F32_16X16X4_F32
    layer55(wn1, b1p, ta1, phi, pdv, lane);
    layer55(wn2, b2p, ta2, phi, pdv, lane);
    // layer 3: 5 -> 1 (in-lane reduction)
    float z  = b3g[dn];
    float pq = 0.0f;
#pragma unroll
    for (int l = 0; l < MH; ++l) {
      float w = wn3[l];
      z  = fmaf(phi[l], w, z);
      pq = fmaf(pdv[l], w, pq);
    }
    float s   = fast_rcp(1.0f + __expf(-z));   // sigmoid: single v_rcp_f32
    float pdf = pq * s * (1.0f - s);
    Lsum += __logf(pdf + 1e-10f);
  }
  if (lane < 16) Lout[mrow * NN + n] = Lsum;
}

// Per-batch-element mixture reduction:
// maxL = max_n L ; f = sum_n exp(L - maxL) * am_n ; out = log(f+1e-10)+maxL
__global__ __launch_bounds__(NN) void finalize_kernel(const float* __restrict__ Lin,
                                                      const float* __restrict__ amn,
                                                      float* __restrict__ out) {
  __shared__ float red[NN];
  int m = blockIdx.x;
  int t = threadIdx.x;
  float Ln = Lin[m * NN + t];
  red[t] = Ln;
  __syncthreads();
  for (int s = NN / 2; s > 0; s >>= 1) {
    if (t < s) red[t] = fmaxf(red[t], red[t + s]);
    __syncthreads();
  }
  float maxL = red[0];
  __syncthreads();
  red[t] = __expf(Ln - maxL) * amn[t];
  __syncthreads();
  for (int s = NN / 2; s > 0; s >>= 1) {
    if (t < s) red[t] += red[t + s];
    __syncthreads();
  }
  if (t == 0) out[m] = __logf(red[0] + 1e-10f) + maxL;
}

// ---------------- host launcher ----------------
extern "C" void kernel_launch(void* const* d_in, const int* in_sizes, int n_in,
                              void* d_out, int out_size, void* d_ws, size_t ws_size,
                              hipStream_t stream) {
  const float* X     = (const float*)d_in[0];
  const float* w0    = (const float*)d_in[1];
  const float* b0    = (const float*)d_in[2];
  const float* a0    = (const float*)d_in[3];
  const float* w1    = (const float*)d_in[4];
  const float* b1    = (const float*)d_in[5];
  const float* a1    = (const float*)d_in[6];
  const float* w2    = (const float*)d_in[7];
  const float* b2    = (const float*)d_in[8];
  const float* a2    = (const float*)d_in[9];
  const float* w3    = (const float*)d_in[10];
  const float* b3    = (const float*)d_in[11];
  const float* a_mix = (const float*)d_in[12];
  float* ws  = (float*)d_ws;
  float* out = (float*)d_out;

  auto xf = [&](const float* s, float* dv, int cnt, int mode) {
    xform_kernel<<<(cnt + 255) / 256, 256, 0, stream>>>(s, dv, cnt, mode);
  };
  xf(w0, ws + O_WN0, DD * NN * MH,      0);
  xf(w1, ws + O_WN1, DD * NN * MH * MH, 0);
  xf(w2, ws + O_WN2, DD * NN * MH * MH, 0);
  xf(w3, ws + O_WN3, DD * NN * MH,      0);
  xf(a0, ws + O_TA0, DD * NN * MH,      1);
  xf(a1, ws + O_TA1, DD * NN * MH,      1);
  xf(a2, ws + O_TA2, DD * NN * MH,      1);
  amix_kernel<<<1, NN, 0, stream>>>(a_mix, ws + O_AM);

  // 512 components * 64 batch tiles = 32768 waves; 8 waves per 256-thread block
  mlp_kernel<<<(NN * (BATCH / 16)) / 8, 256, 0, stream>>>(
      X, b0, b1, b2, b3, ws, ws + O_L);

  finalize_kernel<<<BATCH, NN, 0, stream>>>(ws + O_L, ws + O_AM, out);
}